// NASEERLayer_12455405159016
// MI455X (gfx1250) — compile-verified
//
#include <hip/hip_runtime.h>
#include <math.h>

#define B_ 2
#define T_ 384
#define D_ 512
#define H_ 8
#define HD_ 64
#define K_ 48

typedef __attribute__((ext_vector_type(2))) float v2f;
typedef __attribute__((ext_vector_type(8))) float v8f;

// ---------------------------------------------------------------------------
// Kernel 1: per-token head softmax + superposition + sp projection
// x:(BT,512) g_w:(8,512) sp_w:(512,64) -> x_super:(BT,512)
// ---------------------------------------------------------------------------
__global__ __launch_bounds__(256)
void super_kernel(const float* __restrict__ x,
                  const float* __restrict__ g_w, const float* __restrict__ g_b,
                  const float* __restrict__ sp_w, const float* __restrict__ sp_b,
                  float* __restrict__ x_super)
{
    __shared__ float red[256];
    __shared__ float logits[H_];
    __shared__ float alpha[H_];
    __shared__ float xsh[HD_];
    const int t = blockIdx.x;
    const int tid = threadIdx.x;
    const float* xr = x + (long)t * D_;

    // logits[h] = dot(x[t], g_w[h]) + g_b[h]
    for (int h = 0; h < H_; ++h) {
        float p = xr[tid] * g_w[h * D_ + tid] + xr[tid + 256] * g_w[h * D_ + tid + 256];
        red[tid] = p;
        __syncthreads();
        for (int s = 128; s > 0; s >>= 1) {
            if (tid < s) red[tid] += red[tid + s];
            __syncthreads();
        }
        if (tid == 0) logits[h] = red[0] + g_b[h];
        __syncthreads();
    }
    if (tid == 0) {
        float mx = logits[0];
        for (int h = 1; h < H_; ++h) mx = fmaxf(mx, logits[h]);
        float s = 0.f;
        for (int h = 0; h < H_; ++h) { float e = __expf(logits[h] - mx); alpha[h] = e; s += e; }
        float inv = 1.f / s;
        for (int h = 0; h < H_; ++h) alpha[h] *= inv;
    }
    __syncthreads();
    if (tid < HD_) {
        float s = 0.f;
        for (int h = 0; h < H_; ++h) s += alpha[h] * xr[h * HD_ + tid];
        xsh[tid] = s;
    }
    __syncthreads();
    for (int d = tid; d < D_; d += 256) {
        const float* wr = sp_w + (long)d * HD_;
        float s = sp_b[d];
        #pragma unroll 8
        for (int k = 0; k < HD_; ++k) s += wr[k] * xsh[k];
        x_super[(long)t * D_ + d] = s;
    }
}

// ---------------------------------------------------------------------------
// WMMA f32 GEMM: C[M,N] = A[M,K] @ W[N,K]^T (+bias).  One wave per 16x16 tile.
// MODE 0: out0 = acc + bias
// MODE 1: g = sigmoid(acc + bias); out0 = g; out1 = g*g
// MODE 2: out0 = acc                (bias unused; Gram matrix)
// Batched via blockIdx.y with element strides sA,sW,sO.
// ---------------------------------------------------------------------------
template <int MODE>
__global__ __launch_bounds__(256)
void gemm_nt_wmma(const float* __restrict__ A, const float* __restrict__ W,
                  const float* __restrict__ bias,
                  float* __restrict__ out0, float* __restrict__ out1,
                  int M, int N, int Kd,
                  long sA, long sW, long sO)
{
    const int lane   = threadIdx.x & 31;
    const int wave   = threadIdx.x >> 5;
    const int tilesN = N >> 4;
    const int tile   = blockIdx.x * 8 + wave;
    const int tm     = tile / tilesN;
    const int tn     = tile - tm * tilesN;
    const int m      = lane & 15;     // M-row for A frag, N-col for B/C frags
    const int half   = lane >> 4;

    const float* Ab = A + (long)blockIdx.y * sA;
    const float* Wb = W + (long)blockIdx.y * sW;
    float* O0 = out0 + (long)blockIdx.y * sO;

    // A frag: a[v] = A[tm*16+m][kb + 2*half + v]   (ISA f32 16x4 layout)
    // B frag: b[v] = W[tn*16+n][kb + 2*half + v]   (B = W^T, 4x16 layout)
    const float* arow = Ab + (long)(tm * 16 + m) * Kd + 2 * half;
    const float* wrow = Wb + (long)(tn * 16 + m) * Kd + 2 * half;

    v8f acc = {0.f, 0.f, 0.f, 0.f, 0.f, 0.f, 0.f, 0.f};
    for (int kb = 0; kb < Kd; kb += 4) {
        v2f af = *(const v2f*)(arow + kb);
        v2f bf = *(const v2f*)(wrow + kb);
        acc = __builtin_amdgcn_wmma_f32_16x16x4_f32(
            /*neg_a=*/false, af, /*neg_b=*/false, bf,
            /*c_mod=*/(short)0, acc, /*reuse_a=*/false, /*reuse_b=*/false);
    }

    const int col = tn * 16 + m;
    const float bv = (MODE == 2) ? 0.f : bias[col];
    #pragma unroll
    for (int r = 0; r < 8; ++r) {
        const int row = tm * 16 + r + 8 * half;  // C/D: lanes16-31 hold M=r+8
        const long off = (long)row * N + col;
        float v = acc[r] + bv;
        if (MODE == 1) {
            float g = 1.f / (1.f + __expf(-v));
            O0[off] = g;
            out1[(long)blockIdx.y * sO + off] = g * g;
        } else {
            O0[off] = v;
        }
    }
}

// ---------------------------------------------------------------------------
// Kernel 4: per-token top-48 over Gram row (smaller-index tie-break, matching
// jax.lax.top_k) then entangled = gate_i * sum_k gate_{j_k};
// x_out = x_super + entangled
// ---------------------------------------------------------------------------
__global__ __launch_bounds__(128)
void topk_entangle(const float* __restrict__ S, const float* __restrict__ gate,
                   const float* __restrict__ x_super, float* __restrict__ x_out)
{
    __shared__ float sc[T_];
    __shared__ int   sel[K_];
    __shared__ float rv[128];
    __shared__ int   ri[128];
    const int bt  = blockIdx.x;
    const int b   = bt / T_;
    const int tid = threadIdx.x;
    const float* srow = S + (long)b * T_ * T_ + (long)(bt - b * T_) * T_;

    for (int j = tid; j < T_; j += 128) sc[j] = srow[j];
    __syncthreads();

    for (int k = 0; k < K_; ++k) {
        float bv = -INFINITY; int bi = T_;
        for (int j = tid; j < T_; j += 128) {
            float v = sc[j];
            if (v > bv) { bv = v; bi = j; }
        }
        rv[tid] = bv; ri[tid] = bi;
        __syncthreads();
        for (int s = 64; s > 0; s >>= 1) {
            if (tid < s) {
                float v2 = rv[tid + s]; int i2 = ri[tid + s];
                if (v2 > rv[tid] || (v2 == rv[tid] && i2 < ri[tid])) {
                    rv[tid] = v2; ri[tid] = i2;
                }
            }
            __syncthreads();
        }
        if (tid == 0) { sel[k] = ri[0]; sc[ri[0]] = -INFINITY; }
        __syncthreads();
    }

    const float* gi = gate + (long)bt * D_;
    for (int d = tid; d < D_; d += 128) {
        float s = 0.f;
        for (int k = 0; k < K_; ++k)
            s += gate[((long)b * T_ + sel[k]) * D_ + d];
        x_out[(long)bt * D_ + d] = x_super[(long)bt * D_ + d] + gi[d] * s;
    }
}

// ---------------------------------------------------------------------------
// Kernel 6: LayerNorm over D=512 (256 threads, 2 elems/thread)
// ---------------------------------------------------------------------------
__global__ __launch_bounds__(256)
void ln_kernel(const float* __restrict__ y, const float* __restrict__ g,
               const float* __restrict__ bta, float* __restrict__ out)
{
    __shared__ float red[256];
    __shared__ float smu, srstd;
    const long t = blockIdx.x;
    const int tid = threadIdx.x;
    const float* yr = y + t * D_;
    float v0 = yr[tid], v1 = yr[tid + 256];

    red[tid] = v0 + v1;
    __syncthreads();
    for (int s = 128; s > 0; s >>= 1) { if (tid < s) red[tid] += red[tid + s]; __syncthreads(); }
    if (tid == 0) smu = red[0] * (1.f / D_);
    __syncthreads();
    const float mu = smu;
    const float d0 = v0 - mu, d1 = v1 - mu;
    red[tid] = d0 * d0 + d1 * d1;
    __syncthreads();
    for (int s = 128; s > 0; s >>= 1) { if (tid < s) red[tid] += red[tid + s]; __syncthreads(); }
    if (tid == 0) srstd = rsqrtf(red[0] * (1.f / D_) + 1e-5f);
    __syncthreads();
    const float rs = srstd;
    out[t * D_ + tid]       = g[tid]       * d0 * rs + bta[tid];
    out[t * D_ + tid + 256] = g[tid + 256] * d1 * rs + bta[tid + 256];
}

// ---------------------------------------------------------------------------
extern "C" void kernel_launch(void* const* d_in, const int* in_sizes, int n_in,
                              void* d_out, int out_size, void* d_ws, size_t ws_size,
                              hipStream_t stream)
{
    const float* x      = (const float*)d_in[0];
    const float* g_w    = (const float*)d_in[1];
    const float* g_b    = (const float*)d_in[2];
    const float* sp_w   = (const float*)d_in[3];
    const float* sp_b   = (const float*)d_in[4];
    const float* gate_w = (const float*)d_in[5];
    const float* gate_b = (const float*)d_in[6];
    const float* U_w    = (const float*)d_in[7];
    const float* U_b    = (const float*)d_in[8];
    const float* ln_g   = (const float*)d_in[9];
    const float* ln_b   = (const float*)d_in[10];
    float* out = (float*)d_out;

    const long NT = (long)B_ * T_;          // 768 tokens
    float* ws   = (float*)d_ws;
    float* xs   = ws;                        // (NT, D)
    float* gate = xs + NT * D_;              // (NT, D)
    float* q    = gate + NT * D_;            // (NT, D)  gate^2
    float* S    = q + NT * D_;               // (B, T, T) Gram
    float* xo   = S + (long)B_ * T_ * T_;    // (NT, D)
    float* y    = xo + NT * D_;              // (NT, D)

    // 1) superposition + sp projection
    super_kernel<<<dim3((unsigned)NT), 256, 0, stream>>>(x, g_w, g_b, sp_w, sp_b, xs);

    // 2) gate = sigmoid(xs @ gate_w^T + gate_b), q = gate^2   [WMMA]
    {
        int tiles = (int)(NT / 16) * (D_ / 16);   // 48*32 = 1536
        gemm_nt_wmma<1><<<dim3(tiles / 8, 1), 256, 0, stream>>>(
            xs, gate_w, gate_b, gate, q, (int)NT, D_, D_, 0, 0, 0);
    }

    // 3) S[b] = q[b] @ q[b]^T   (batched Gram)                 [WMMA]
    {
        int tiles = (T_ / 16) * (T_ / 16);        // 24*24 = 576
        gemm_nt_wmma<2><<<dim3(tiles / 8, B_), 256, 0, stream>>>(
            q, q, nullptr, S, nullptr, T_, T_, D_,
            (long)T_ * D_, (long)T_ * D_, (long)T_ * T_);
    }

    // 4) top-48 + entangle
    topk_entangle<<<dim3((unsigned)NT), 128, 0, stream>>>(S, gate, xs, xo);

    // 5) y = xo @ U_w^T + U_b                                   [WMMA]
    {
        int tiles = (int)(NT / 16) * (D_ / 16);
        gemm_nt_wmma<0><<<dim3(tiles / 8, 1), 256, 0, stream>>>(
            xo, U_w, U_b, y, nullptr, (int)NT, D_, D_, 0, 0, 0);
    }

    // 6) LayerNorm
    ln_kernel<<<dim3((unsigned)NT), 256, 0, stream>>>(y, ln_g, ln_b, out);
}